// QuadricBaseNN_59880434040939
// MI455X (gfx1250) — compile-verified
//
#include <hip/hip_runtime.h>
#include <hip/hip_bf16.h>

// ---------------------------------------------------------------------------
// QuadricBaseNN on MI455X (gfx1250, wave32).
//
// Phase 1 (init):   zero accumulators; pre-swizzle B-fragments
//                   B[:,m] = (-2px,-2py,-2pz,||p||^2) into per-lane WMMA layout.
// Phase 2 (cluster): argmin over M=1024 clusters via V_WMMA_F32_16X16X4_F32
//                   (score = [s,1]·[-2p,pp2]^T, K=4 exactly). Inner loop is
//                   software-pipelined 2 tiles deep so the ds_load_b64 latency
//                   hides behind 4 WMMAs + min/idx tracking. Branchless
//                   shfl_xor butterfly finishes the argmin; then 14 global f32
//                   atomics/sample (upper-tri quadric + normal + count).
// Phase 3 (solve):  per-cluster closed-form symmetric 3x3 eigensolve +
//                   thresholded pseudo-inverse, write all 4 outputs.
// ---------------------------------------------------------------------------

#define BATCH   4
#define MCLUST  1024
#define NSAMP   65536
#define TILES   (MCLUST / 16)          // 64 WMMA column tiles
#define ACC_F   (BATCH * MCLUST * 14)  // 57344 floats of atomics accumulator
#define BSWZ_F2 (BATCH * TILES * 32)   // 8192 float2 swizzled B fragments
#define EPSV    0.01f

typedef float v2f __attribute__((ext_vector_type(2)));
typedef float v8f __attribute__((ext_vector_type(8)));

// ---------------------------------------------------------------------------
// Kernel 0: zero accumulator + build per-lane swizzled WMMA B fragments.
// B-matrix 4x16 layout (mirror of A 16x4 ISA table): lane = column (m%16),
// VGPR0 = K0 (lanes 0-15) / K2 (lanes 16-31), VGPR1 = K1 / K3.
// K = (-2px, -2py, -2pz, pp2)  =>  lane<16: (x=-2px, y=-2py)
//                                  lane>=16:(x=-2pz, y= pp2)
// ---------------------------------------------------------------------------
__global__ void qb_init_kernel(const float* __restrict__ pred,
                               float* __restrict__ acc,
                               float2* __restrict__ bswz) {
    const int i = blockIdx.x * blockDim.x + threadIdx.x;
    if (i < ACC_F) acc[i] = 0.0f;
    if (i < BSWZ_F2) {
        const int lane = i & 31;
        const int t    = (i >> 5) & (TILES - 1);
        const int b    = i / (TILES * 32);
        const int m    = t * 16 + (lane & 15);
        const float* p = pred + ((size_t)(b * MCLUST + m)) * 3;
        const float px = p[0], py = p[1], pz = p[2];
        float2 v;
        if (lane < 16) { v.x = -2.0f * px; v.y = -2.0f * py; }
        else           { v.x = -2.0f * pz; v.y = px*px + py*py + pz*pz; }
        bswz[i] = v;
    }
}

// Branchless (score, idx) argmin combine with first-occurrence tie-break.
__device__ __forceinline__ void qb_take(float& mnv, int& ixv, float om, int oi) {
    const bool t = (om < mnv) | ((om == mnv) & (oi < ixv));
    mnv = t ? om : mnv;
    ixv = t ? oi : ixv;
}

// ---------------------------------------------------------------------------
// Kernel 1: WMMA argmin + quadric scatter.
// 1 block = 8 waves, each wave owns 32 samples (two 16-row WMMA A groups).
// 256 blocks per batch image => grid = 1024 blocks.
// ---------------------------------------------------------------------------
__global__ __launch_bounds__(256) void qb_cluster_kernel(
    const float* __restrict__ samples, const float* __restrict__ normals,
    const float2* __restrict__ bswz, float* __restrict__ acc) {

    __shared__ float2 sB[TILES * 32];   // 16 KB: swizzled B frags, batch-local
    __shared__ int    sIdx[8][32];      // per-wave argmin results

    const int lane  = threadIdx.x & 31;
    const int wave  = threadIdx.x >> 5;
    const int chunk = blockIdx.x * 8 + wave;       // 32-sample chunk id
    const int cPerB = NSAMP / 32;                   // 2048 (block-uniform batch)
    const int b     = chunk / cPerB;
    const int s0    = (chunk % cPerB) * 32;

    // Stage this batch's B fragments into LDS (coalesced, reused by 8 waves).
    const float2* gB = bswz + (size_t)b * (TILES * 32);
    for (int i = threadIdx.x; i < TILES * 32; i += 256) sB[i] = gB[i];
    __syncthreads();

    // A fragments: 16x4 layout -> lane holds row (s0 + lane%16);
    // VGPR0 = K0 (lanes 0-15) / K2 (lanes 16-31), VGPR1 = K1 / K3.
    // A row = (sx, sy, sz, 1).
    const float* smp = samples + (size_t)b * NSAMP * 3;
    const int  sA  = s0 + (lane & 15);
    const bool hi  = lane >= 16;
    const float sx0 = smp[sA * 3 + 0],        sy0 = smp[sA * 3 + 1],        sz0 = smp[sA * 3 + 2];
    const float sx1 = smp[(sA + 16) * 3 + 0], sy1 = smp[(sA + 16) * 3 + 1], sz1 = smp[(sA + 16) * 3 + 2];
    v2f a0, a1;
    a0.x = hi ? sz0 : sx0;  a0.y = hi ? 1.0f : sy0;
    a1.x = hi ? sz1 : sx1;  a1.y = hi ? 1.0f : sy1;

    float mn0[8], mn1[8];
    int   ix0[8], ix1[8];
#pragma unroll
    for (int r = 0; r < 8; ++r) {
        mn0[r] = __builtin_inff(); mn1[r] = __builtin_inff();
        ix0[r] = 0; ix1[r] = 0;
    }
    v8f cz = {};

    // Hot loop: 2-tile software pipeline. Per iteration: 2 ds_load_b64
    // (prefetch t+2/t+3) + 4 WMMA + dual-issued min/idx selects.
    float2 bfA = sB[lane];
    float2 bfB = sB[32 + lane];
    const int colBase = lane & 15;
    for (int t = 0; t < TILES; t += 2) {
        v2f bva; bva.x = bfA.x; bva.y = bfA.y;
        v2f bvb; bvb.x = bfB.x; bvb.y = bfB.y;
        if (t + 2 < TILES) {
            bfA = sB[(t + 2) * 32 + lane];
            bfB = sB[(t + 3) * 32 + lane];
        }
        v8f d0 = __builtin_amdgcn_wmma_f32_16x16x4_f32(false, a0, false, bva,
                                                       (short)0, cz, false, false);
        v8f d1 = __builtin_amdgcn_wmma_f32_16x16x4_f32(false, a1, false, bva,
                                                       (short)0, cz, false, false);
        v8f e0 = __builtin_amdgcn_wmma_f32_16x16x4_f32(false, a0, false, bvb,
                                                       (short)0, cz, false, false);
        v8f e1 = __builtin_amdgcn_wmma_f32_16x16x4_f32(false, a1, false, bvb,
                                                       (short)0, cz, false, false);
        const int colA = (t << 4) + colBase;
        const int colB = colA + 16;
#pragma unroll
        for (int r = 0; r < 8; ++r) {
            // per-lane sequence of columns is strictly increasing, so plain
            // '<' keeps the earliest index automatically
            if (d0[r] < mn0[r]) { mn0[r] = d0[r]; ix0[r] = colA; }
            if (d1[r] < mn1[r]) { mn1[r] = d1[r]; ix1[r] = colA; }
            if (e0[r] < mn0[r]) { mn0[r] = e0[r]; ix0[r] = colB; }
            if (e1[r] < mn1[r]) { mn1[r] = e1[r]; ix1[r] = colB; }
        }
    }

    // Cross-lane argmin: C/D layout -> lane = column, halves hold rows 0-7/8-15.
    // Branchless butterfly over each 16-lane half; tie-break to lowest index
    // (argmin first-occurrence semantics).
#pragma unroll
    for (int xm = 1; xm < 16; xm <<= 1) {
#pragma unroll
        for (int r = 0; r < 8; ++r) {
            float om0 = __shfl_xor(mn0[r], xm, 32);
            int   oi0 = __shfl_xor(ix0[r], xm, 32);
            float om1 = __shfl_xor(mn1[r], xm, 32);
            int   oi1 = __shfl_xor(ix1[r], xm, 32);
            qb_take(mn0[r], ix0[r], om0, oi0);
            qb_take(mn1[r], ix1[r], om1, oi1);
        }
    }

    // lane 0 has rows 0-7, lane 16 has rows 8-15 (per group). Park in LDS so
    // every lane can pick up one sample for the scatter phase.
    if (lane == 0) {
#pragma unroll
        for (int r = 0; r < 8; ++r) { sIdx[wave][r]      = ix0[r]; sIdx[wave][16 + r] = ix1[r]; }
    }
    if (lane == 16) {
#pragma unroll
        for (int r = 0; r < 8; ++r) { sIdx[wave][8 + r]  = ix0[r]; sIdx[wave][24 + r] = ix1[r]; }
    }
    __builtin_amdgcn_wave_barrier();   // intra-wave LDS ordering fence
    __syncthreads();

    // Scatter phase: each lane owns one sample -> 14 f32 atomics into L2-
    // resident accumulator (upper-tri 4x4 quadric + normal + count).
    const int   m  = sIdx[wave][lane];
    const int   s  = s0 + lane;
    const float sx = smp[s * 3 + 0], sy = smp[s * 3 + 1], sz = smp[s * 3 + 2];
    const float* nrm = normals + (size_t)b * NSAMP * 3;
    const float nx = nrm[s * 3 + 0], ny = nrm[s * 3 + 1], nz = nrm[s * 3 + 2];
    const float dp = -(nx * sx + ny * sy + nz * sz);

    float* ab = acc + ((size_t)(b * MCLUST + m)) * 14;
    unsafeAtomicAdd(ab + 0,  nx * nx);
    unsafeAtomicAdd(ab + 1,  nx * ny);
    unsafeAtomicAdd(ab + 2,  nx * nz);
    unsafeAtomicAdd(ab + 3,  nx * dp);
    unsafeAtomicAdd(ab + 4,  ny * ny);
    unsafeAtomicAdd(ab + 5,  ny * nz);
    unsafeAtomicAdd(ab + 6,  ny * dp);
    unsafeAtomicAdd(ab + 7,  nz * nz);
    unsafeAtomicAdd(ab + 8,  nz * dp);
    unsafeAtomicAdd(ab + 9,  dp * dp);
    unsafeAtomicAdd(ab + 10, nx);
    unsafeAtomicAdd(ab + 11, ny);
    unsafeAtomicAdd(ab + 12, nz);
    unsafeAtomicAdd(ab + 13, 1.0f);
}

// ---------------------------------------------------------------------------
// Kernel 2: per-cluster solve. A is PSD symmetric => SVD == eigendecomposition
// (descending). Eigenvector sign is irrelevant (v appears quadratically).
// ---------------------------------------------------------------------------
__device__ __forceinline__ void qb_eigvec(float S00, float S01, float S02,
                                          float S11, float S12, float S22,
                                          float lam, float& vx, float& vy, float& vz) {
    const float r0x = S00 - lam, r0y = S01,       r0z = S02;
    const float r1x = S01,       r1y = S11 - lam, r1z = S12;
    const float r2x = S02,       r2y = S12,       r2z = S22 - lam;
    float c0x = r0y * r1z - r0z * r1y, c0y = r0z * r1x - r0x * r1z, c0z = r0x * r1y - r0y * r1x;
    float c1x = r0y * r2z - r0z * r2y, c1y = r0z * r2x - r0x * r2z, c1z = r0x * r2y - r0y * r2x;
    float c2x = r1y * r2z - r1z * r2y, c2y = r1z * r2x - r1x * r2z, c2z = r1x * r2y - r1y * r2x;
    const float n0 = c0x * c0x + c0y * c0y + c0z * c0z;
    const float n1 = c1x * c1x + c1y * c1y + c1z * c1z;
    const float n2 = c2x * c2x + c2y * c2y + c2z * c2z;
    float bx = c0x, by = c0y, bz = c0z, bn = n0;
    if (n1 > bn) { bx = c1x; by = c1y; bz = c1z; bn = n1; }
    if (n2 > bn) { bx = c2x; by = c2y; bz = c2z; bn = n2; }
    if (bn < 1e-30f) { vx = 1.0f; vy = 0.0f; vz = 0.0f; return; }
    const float inv = rsqrtf(bn);
    vx = bx * inv; vy = by * inv; vz = bz * inv;
}

__global__ __launch_bounds__(128) void qb_solve_kernel(
    const float* __restrict__ pred, const float* __restrict__ acc,
    float* __restrict__ out) {
    const int i = blockIdx.x * blockDim.x + threadIdx.x;
    if (i >= BATCH * MCLUST) return;

    const float* a = acc + (size_t)i * 14;
    const float cnt   = a[13];
    const float inv   = 1.0f / fmaxf(cnt, 1.0f);
    const float A00 = a[0] * inv, A01 = a[1] * inv, A02 = a[2] * inv;
    const float A11 = a[4] * inv, A12 = a[5] * inv, A22 = a[7] * inv;
    const float bx  = -a[3] * inv, by = -a[6] * inv, bz = -a[8] * inv;
    const float mnx = a[10] * inv, mny = a[11] * inv, mnz = a[12] * inv;
    const bool  nonVoid = (mnx * mnx + mny * mny + mnz * mnz) > 0.0f;

    // A_safe = non_void ? A : diag(3,2,1)
    float S00, S01, S02, S11, S12, S22;
    if (nonVoid) { S00 = A00; S01 = A01; S02 = A02; S11 = A11; S12 = A12; S22 = A22; }
    else         { S00 = 3.0f; S01 = 0.0f; S02 = 0.0f; S11 = 2.0f; S12 = 0.0f; S22 = 1.0f; }

    // Closed-form symmetric 3x3 eigenvalues (trigonometric), descending.
    const float tr = S00 + S11 + S22;
    const float qm = tr * (1.0f / 3.0f);
    const float p1 = S01 * S01 + S02 * S02 + S12 * S12;
    const float e0d = S00 - qm, e1d = S11 - qm, e2d = S22 - qm;
    const float p2 = e0d * e0d + e1d * e1d + e2d * e2d + 2.0f * p1;
    const float pp = sqrtf(p2 * (1.0f / 6.0f));

    float l0, l1, l2;
    float v0x, v0y, v0z, v1x, v1y, v1z, v2x, v2y, v2z;
    if (pp < 1e-12f) {
        l0 = l1 = l2 = qm;
        v0x = 1.0f; v0y = 0.0f; v0z = 0.0f;
        v1x = 0.0f; v1y = 1.0f; v1z = 0.0f;
        v2x = 0.0f; v2y = 0.0f; v2z = 1.0f;
    } else {
        const float ip  = 1.0f / pp;
        const float B00 = e0d * ip, B01 = S01 * ip, B02 = S02 * ip;
        const float B11 = e1d * ip, B12 = S12 * ip, B22 = e2d * ip;
        float detB = B00 * (B11 * B22 - B12 * B12)
                   - B01 * (B01 * B22 - B12 * B02)
                   + B02 * (B01 * B12 - B11 * B02);
        float r = fminf(1.0f, fmaxf(-1.0f, 0.5f * detB));
        const float phi = acosf(r) * (1.0f / 3.0f);
        l0 = qm + 2.0f * pp * cosf(phi);
        l2 = qm + 2.0f * pp * cosf(phi + 2.0943951023931953f); // +2pi/3
        l1 = tr - l0 - l2;
        qb_eigvec(S00, S01, S02, S11, S12, S22, l0, v0x, v0y, v0z);
        qb_eigvec(S00, S01, S02, S11, S12, S22, l2, v2x, v2y, v2z);
        v1x = v2y * v0z - v2z * v0y;
        v1y = v2z * v0x - v2x * v0z;
        v1z = v2x * v0y - v2y * v0x;
    }

    // Thresholded pseudo-inverse: mask = (eig / max(eig0,1e-20)) > EPS.
    const float thr = EPSV * fmaxf(l0, 1e-20f);
    const float w0 = (l0 > thr) ? (1.0f / l0) : 0.0f;
    const float w1 = (l1 > thr) ? (1.0f / l1) : 0.0f;
    const float w2 = (l2 > thr) ? (1.0f / l2) : 0.0f;

    // rhs = b - A * p   (original A, per reference)
    const float px = pred[i * 3 + 0], py = pred[i * 3 + 1], pz = pred[i * 3 + 2];
    const float rx = bx - (A00 * px + A01 * py + A02 * pz);
    const float ry = by - (A01 * px + A11 * py + A12 * pz);
    const float rz = bz - (A02 * px + A12 * py + A22 * pz);
    const float t0 = v0x * rx + v0y * ry + v0z * rz;
    const float t1 = v1x * rx + v1y * ry + v1z * rz;
    const float t2 = v2x * rx + v2y * ry + v2z * rz;
    const float dx = w0 * t0 * v0x + w1 * t1 * v1x + w2 * t2 * v2x;
    const float dy = w0 * t0 * v0y + w1 * t1 * v1y + w2 * t2 * v2y;
    const float dz = w0 * t0 * v0z + w1 * t1 * v1z + w2 * t2 * v2z;

    // Outputs (concatenated): A (B,M,3,3) | mean_normals | vstars | non_void
    float* oA = out + (size_t)i * 9;
    oA[0] = A00; oA[1] = A01; oA[2] = A02;
    oA[3] = A01; oA[4] = A11; oA[5] = A12;
    oA[6] = A02; oA[7] = A12; oA[8] = A22;
    float* oN = out + BATCH * MCLUST * 9 + (size_t)i * 3;
    oN[0] = mnx; oN[1] = mny; oN[2] = mnz;
    float* oV = out + BATCH * MCLUST * 12 + (size_t)i * 3;
    oV[0] = nonVoid ? (px + dx) : 0.0f;
    oV[1] = nonVoid ? (py + dy) : 0.0f;
    oV[2] = nonVoid ? (pz + dz) : 0.0f;
    out[BATCH * MCLUST * 15 + i] = nonVoid ? 1.0f : 0.0f;
}

extern "C" void kernel_launch(void* const* d_in, const int* in_sizes, int n_in,
                              void* d_out, int out_size, void* d_ws, size_t ws_size,
                              hipStream_t stream) {
    const float* pred    = (const float*)d_in[0];
    const float* samples = (const float*)d_in[1];
    const float* normals = (const float*)d_in[2];
    float*  out  = (float*)d_out;
    float*  acc  = (float*)d_ws;                                    // 229 KB
    float2* bswz = (float2*)((char*)d_ws + (size_t)ACC_F * sizeof(float)); // 64 KB

    qb_init_kernel<<<(ACC_F + 255) / 256, 256, 0, stream>>>(pred, acc, bswz);
    qb_cluster_kernel<<<(BATCH * NSAMP) / (8 * 32), 256, 0, stream>>>(samples, normals, bswz, acc);
    qb_solve_kernel<<<(BATCH * MCLUST + 127) / 128, 128, 0, stream>>>(pred, acc, out);
}